// Model_2619930051518
// MI455X (gfx1250) — compile-verified
//
#include <hip/hip_runtime.h>

// Problem constants (match reference)
constexpr int kB      = 64;
constexpr int kTopK   = 2;
constexpr int kE      = 16;
constexpr int kC      = 1024;
constexpr int kK      = 4096;
constexpr int kSlots  = kB * kTopK;   // 128 token-slots
constexpr int kRowT   = 16;           // WMMA M tile
constexpr int kCTile  = 128;          // columns per block (8 waves x 16)
constexpr int kKStep  = 32;           // k-chunk staged in LDS per iteration
constexpr int kLdsStr = kKStep + 4;   // 36 floats: rows 16B-aligned (144B), conflict-free
constexpr int kIters  = kK / kKStep;  // 128

typedef float v2f __attribute__((ext_vector_type(2)));
typedef float v8f __attribute__((ext_vector_type(8)));

// CDNA5 async global->LDS copies (ASYNCcnt-tracked, no VGPR round trip).
#define ASYNC_LDS_B128(ldsaddr, gptr)                                          \
    asm volatile("global_load_async_to_lds_b128 %0, %1, off"                   \
                 :: "v"(ldsaddr), "v"(gptr) : "memory")
#define ASYNC_LDS_B64(ldsaddr, gptr)                                           \
    asm volatile("global_load_async_to_lds_b64 %0, %1, off"                    \
                 :: "v"(ldsaddr), "v"(gptr) : "memory")
#define WAIT_ASYNC_LE5() asm volatile("s_wait_asynccnt 5" ::: "memory")
#define WAIT_ASYNC_0()   asm volatile("s_wait_asynccnt 0" ::: "memory")

// ---------------------------------------------------------------------------
// Pass 0: bucket the 128 (b,slot) pairs by expert. Single thread => fully
// deterministic list order.
// ---------------------------------------------------------------------------
__global__ void k_build_lists(const int* __restrict__ eidx,
                              int* __restrict__ counts,
                              int* __restrict__ lists) {
    if (blockIdx.x != 0 || threadIdx.x != 0) return;
    int cnt[kE];
#pragma unroll
    for (int e = 0; e < kE; ++e) cnt[e] = 0;
    for (int t = 0; t < kSlots; ++t) {
        const int e = eidx[t] & (kE - 1);
        lists[e * kSlots + cnt[e]] = t;   // t = b*TOPK + slot
        ++cnt[e];
    }
#pragma unroll
    for (int e = 0; e < kE; ++e) counts[e] = cnt[e];
}

// ---------------------------------------------------------------------------
// Pass 1: per-expert grouped GEMM with double-buffered async LDS staging.
// One block = 16-token row tile x 128-column tile for one expert.
// ---------------------------------------------------------------------------
__launch_bounds__(256, 2)
__global__ void k_moe_gemm(const float* __restrict__ act,    // [slots, K]
                           const float* __restrict__ wts,    // [E, C, K]
                           const float* __restrict__ bias,   // [E, C]
                           const int*   __restrict__ counts, // [E]
                           const int*   __restrict__ lists,  // [E, slots]
                           float*       __restrict__ tbuf) { // [slots, C]
    const int e     = blockIdx.z;
    const int row0  = blockIdx.y * kRowT;
    const int cbase = blockIdx.x * kCTile;
    const int M     = counts[e];
    if (row0 >= M) return;  // uniform per-block: EXEC stays all-ones for WMMA

    __shared__ float lds_w[2][kCTile][kLdsStr];  // [buf][c_local][k_local]
    __shared__ float lds_a[2][kRowT][kLdsStr];   // [buf][row][k_local]
    __shared__ int   tok[kRowT];

    const int tid  = threadIdx.x;
    const int lane = tid & 31;
    const int wave = tid >> 5;

    if (tid < kRowT) {
        const int r = row0 + tid;
        tok[tid] = (r < M) ? lists[e * kSlots + r] : -1;
    }
    __syncthreads();

    const float* __restrict__ wbase = wts + (size_t)e * kC * kK;

    // W staging: thread covers column wrow + j*32, 16B chunk at wq.
    // Lanes 0..7 form one contiguous 128B k-row of a column => fully coalesced.
    const int wrow = tid >> 3;            // 0..31
    const int wq   = (tid & 7) * 4;       // 0,4,..,28
    const float* __restrict__ wsrc = wbase + (size_t)(cbase + wrow) * kK + wq;

    // A staging: 256 threads x float2 = 16 rows x 32 k. Padding rows read row
    // tok fallback 0 (harmless: their WMMA outputs are never stored).
    const int arow = tid >> 4;            // 0..15
    const int akp  = (tid & 15) * 2;      // 0,2,..,30
    const int atok = max(tok[arow], 0);
    const float* __restrict__ asrc = act + (size_t)atok * kK + akp;

    // Issue all 5 async copies for chunk starting at k0 into buffer `buf`.
    auto issue_chunk = [&](int k0, int buf) {
#pragma unroll
        for (int j = 0; j < 4; ++j) {
            unsigned wdst = (unsigned)(unsigned long long)&lds_w[buf][wrow + j * 32][wq];
            ASYNC_LDS_B128(wdst, (const void*)(wsrc + (size_t)j * 32 * kK + k0));
        }
        unsigned adst = (unsigned)(unsigned long long)&lds_a[buf][arow][akp];
        ASYNC_LDS_B64(adst, (const void*)(asrc + k0));
    };

    v8f acc = {};

    issue_chunk(0, 0);

    for (int it = 0; it < kIters; ++it) {
        const int cur = it & 1;
        if (it + 1 < kIters) {
            issue_chunk((it + 1) * kKStep, cur ^ 1);
            WAIT_ASYNC_LE5();   // in-order completion: only next chunk in flight
        } else {
            WAIT_ASYNC_0();
        }
        __syncthreads();        // all waves' copies for `cur` have landed

        // ---- 8 x V_WMMA_F32_16X16X4_F32; wave owns cols [wave*16, +16) ----
        // A 16x4: lanes 0-15 M=lane K={0,1}; lanes 16-31 M=lane-16 K={2,3}
        // B 4x16: mirrored (lanes 0-15 N=lane K={0,1}; 16-31 K={2,3})
        const int n     = lane & 15;
        const int khalf = (lane >= 16) ? 2 : 0;
#pragma unroll
        for (int kk = 0; kk < kKStep / 4; ++kk) {
            const int koff = kk * 4 + khalf;
            v2f afrag = *(const v2f*)&lds_a[cur][n][koff];
            v2f bfrag = *(const v2f*)&lds_w[cur][wave * 16 + n][koff];
            acc = __builtin_amdgcn_wmma_f32_16x16x4_f32(
                false, afrag, false, bfrag, (short)0, acc, false, false);
        }
        __syncthreads();        // done reading `cur` before it is re-filled
    }

    // ---- epilogue: + bias, scatter rows to t_buf ----
    // D layout: VGPR v holds (M=v, N=lane&15) lanes 0-15 and (M=v+8) lanes 16-31.
    const int   c     = cbase + wave * 16 + (lane & 15);
    const float bv    = bias[e * kC + c];
    const int   mbase = (lane >= 16) ? 8 : 0;
#pragma unroll
    for (int v = 0; v < 8; ++v) {
        const int m = mbase + v;
        if (row0 + m < M) {
            const int t = tok[m];
            tbuf[(size_t)t * kC + c] = acc[v] + bv;
        }
    }
}

// ---------------------------------------------------------------------------
// Pass 2: out = residual + ew0*t0 + ew1*t1
// ---------------------------------------------------------------------------
__global__ void k_combine(const float* __restrict__ resid,
                          const float* __restrict__ ew,
                          const float* __restrict__ tbuf,
                          float* __restrict__ out) {
    const int i = blockIdx.x * blockDim.x + threadIdx.x;
    if (i >= kB * kC) return;
    const int b = i >> 10;          // / kC
    const int c = i & (kC - 1);
    const float t0 = tbuf[(size_t)(b * kTopK + 0) * kC + c];
    const float t1 = tbuf[(size_t)(b * kTopK + 1) * kC + c];
    out[i] = fmaf(ew[b * kTopK + 1], t1,
             fmaf(ew[b * kTopK + 0], t0, resid[i]));
}

// ---------------------------------------------------------------------------
extern "C" void kernel_launch(void* const* d_in, const int* in_sizes, int n_in,
                              void* d_out, int out_size, void* d_ws, size_t ws_size,
                              hipStream_t stream) {
    (void)in_sizes; (void)n_in; (void)out_size; (void)ws_size;
    const float* act   = (const float*)d_in[0];  // [B, TOPK, K]
    const int*   eidx  = (const int*)  d_in[1];  // [B, TOPK] int32
    const float* ew    = (const float*)d_in[2];  // [B, TOPK]
    const float* w2    = (const float*)d_in[3];  // [E, C, K]
    const float* b2    = (const float*)d_in[4];  // [E, C]
    const float* resid = (const float*)d_in[5];  // [B, C]
    float* out = (float*)d_out;

    // Workspace: [counts 16 ints][lists 16*128 ints] ... @16KB: tbuf 128*1024 f32
    int*   counts = (int*)d_ws;
    int*   lists  = counts + kE;
    float* tbuf   = (float*)((char*)d_ws + (16u << 10));

    k_build_lists<<<1, 32, 0, stream>>>(eidx, counts, lists);

    dim3 grid(kC / kCTile, kSlots / kRowT, kE);  // (8, 8, 16)
    k_moe_gemm<<<grid, 256, 0, stream>>>(act, w2, b2, counts, lists, tbuf);

    k_combine<<<(kB * kC + 255) / 256, 256, 0, stream>>>(resid, ew, tbuf, out);
}